// RNNDecoder_17085379904389
// MI455X (gfx1250) — compile-verified
//
#include <hip/hip_runtime.h>
#include <cmath>

// ---------------------------------------------------------------------------
// RNN greedy decoder for MI455X (gfx1250, wave32, WMMA).
// B=128, D_IN=E=H=512, V=32000, L_MSG=16, N_LAYERS=2.
// GEMMs on V_WMMA_F32_16X16X4_F32 (full f32 -> preserves argmax semantics).
// Block = 8 waves = full M=128 x N=64 tile; shared W chunk staged in LDS
// (double-buffered, bank-conflict-free) so weights are read from L2 once
// per step instead of 8x.
// ---------------------------------------------------------------------------

typedef float v2f __attribute__((ext_vector_type(2)));
typedef float v8f __attribute__((ext_vector_type(8)));

#define WMMA_F32_16x16x4(a, b, c) \
    __builtin_amdgcn_wmma_f32_16x16x4_f32(false, (a), false, (b), (short)0, (c), false, false)

#define KC 32            // K-chunk staged in LDS per buffer
#define LDS_STRIDE 36    // 144B rows: 16B aligned (b128 stores) + conflict-free b64 reads

// One K-accumulation pass: acc += X[128xK-tile] * W[64rows x K]^T using LDS-staged W.
__device__ __forceinline__ void gemm_pass(
    const float* __restrict__ X, const float* __restrict__ W, int K,
    v8f acc[4], float* __restrict__ lds,   // lds: 2 buffers of 64*LDS_STRIDE floats
    int m0, int n0, int lm, int half, int srow, int scol)
{
    const float* xrow = X + (long)(m0 + lm) * K;
    const float* wsrc = W + (long)(n0 + srow) * K + scol;  // this thread's staging src
    const int nchunks = K / KC;

    // Prologue: stage chunk 0 into buffer 0 (coalesced 2x b128 per thread).
    {
        float4 r0 = *(const float4*)(wsrc);
        float4 r1 = *(const float4*)(wsrc + 4);
        float* dst = lds + srow * LDS_STRIDE + scol;
        *(float4*)(dst)     = r0;
        *(float4*)(dst + 4) = r1;
    }
    __syncthreads();

    for (int c = 0; c < nchunks; ++c) {
        const int buf = c & 1;

        // Prefetch next W chunk into registers while computing this one.
        float4 p0, p1;
        const bool more = (c + 1) < nchunks;
        if (more) {
            const float* src = wsrc + (c + 1) * KC;
            p0 = *(const float4*)(src);
            p1 = *(const float4*)(src + 4);
        }

        // Compute: 8 k-steps of 4, 4 WMMAs each, B-fragments from LDS.
        const float* xk = xrow + c * KC;
        const float* lb = lds + buf * (64 * LDS_STRIDE) + lm * LDS_STRIDE;
#pragma unroll
        for (int k0 = 0; k0 < KC; k0 += 4) {
            const int k = k0 + (half << 1);
            v2f a  = *(const v2f*)(xk + k);
            v2f b0 = *(const v2f*)(lb + 0 * 16 * LDS_STRIDE + k);
            v2f b1 = *(const v2f*)(lb + 1 * 16 * LDS_STRIDE + k);
            v2f b2 = *(const v2f*)(lb + 2 * 16 * LDS_STRIDE + k);
            v2f b3 = *(const v2f*)(lb + 3 * 16 * LDS_STRIDE + k);
            acc[0] = WMMA_F32_16x16x4(a, b0, acc[0]);
            acc[1] = WMMA_F32_16x16x4(a, b1, acc[1]);
            acc[2] = WMMA_F32_16x16x4(a, b2, acc[2]);
            acc[3] = WMMA_F32_16x16x4(a, b3, acc[3]);
        }
        __syncthreads();

        if (more) {
            float* dst = lds + (1 - buf) * (64 * LDS_STRIDE) + srow * LDS_STRIDE + scol;
            *(float4*)(dst)     = p0;
            *(float4*)(dst + 4) = p1;
        }
        __syncthreads();
    }
}

// out[128][N] = X[128][K]*W[N][K]^T (+ X2[128][K2]*W2[N][K2]^T) + bias (+bias2), opt tanh.
// grid.x = N/64 ; block = 256 threads (8 waves); wave w owns M rows [16w,16w+16).
__global__ __launch_bounds__(256) void gemm_xwT_wmma(
    const float* __restrict__ X,  const float* __restrict__ W,
    const float* __restrict__ bias,
    const float* __restrict__ X2, const float* __restrict__ W2,
    const float* __restrict__ bias2,
    float* __restrict__ out, long out_stride,
    int N, int K, int K2, int do_tanh)
{
    __shared__ float ldsW[2 * 64 * LDS_STRIDE];

    const int tid  = threadIdx.x;
    const int wave = tid >> 5;
    const int lane = tid & 31;
    const int half = lane >> 4;       // 0: K pair {0,1}/rows 0-7 ; 1: K pair {2,3}/rows 8-15
    const int lm   = lane & 15;       // M index (A) / N index (B,C)
    const int srow = tid >> 2;        // staging: 4 threads per W row
    const int scol = (tid & 3) * 8;   // staging: 8 floats (2x b128) per thread
    const int n0   = blockIdx.x << 6;
    const int m0   = wave << 4;

    v8f acc[4] = {v8f{}, v8f{}, v8f{}, v8f{}};

    gemm_pass(X, W, K, acc, ldsW, m0, n0, lm, half, srow, scol);
    if (X2 != nullptr) {
        gemm_pass(X2, W2, K2, acc, ldsW, m0, n0, lm, half, srow, scol);
    }

    // Epilogue per ISA C/D layout: VGPR r of lane (half,lm) holds
    // element [m0 + r + 8*half][n0 + j*16 + lm].
#pragma unroll
    for (int j = 0; j < 4; ++j) {
        const int n = n0 + j * 16 + lm;
        float bj = bias[n];
        if (bias2 != nullptr) bj += bias2[n];
#pragma unroll
        for (int r = 0; r < 8; ++r) {
            const int m = m0 + r + (half << 3);
            float v = acc[j][r] + bj;
            if (do_tanh) v = tanhf(v);
            out[(long)m * out_stride + n] = v;
        }
    }
}

// x[b][e] = sos[e] for all b
__global__ void bcast_sos_kernel(const float* __restrict__ sos,
                                 float* __restrict__ x, int total, int E)
{
    int i = blockIdx.x * blockDim.x + threadIdx.x;
    if (i < total) x[i] = sos[i % E];
}

// Per batch row: argmax over V logits (first-occurrence tie-break like jnp.argmax),
// write symbol (as float) into d_out, gather embed_table[sym] into x_next.
__global__ __launch_bounds__(256) void argmax_embed_kernel(
    const float* __restrict__ logits, long row_stride,   // logits + t*V ; stride L*V
    float* __restrict__ symbols, int t, int L, int V,
    const float* __restrict__ embed, int E,
    float* __restrict__ xout)
{
    const int b = blockIdx.x;
    const float* row = logits + (long)b * row_stride;

    float bestv = -__builtin_inff();
    int   besti = 0;
    for (int v = threadIdx.x; v < V; v += 256) {
        float val = row[v];
        if (val > bestv) { bestv = val; besti = v; }
    }

    __shared__ float sv[256];
    __shared__ int   si[256];
    sv[threadIdx.x] = bestv;
    si[threadIdx.x] = besti;
    __syncthreads();
    for (int s = 128; s > 0; s >>= 1) {
        if (threadIdx.x < s) {
            float ov = sv[threadIdx.x + s];
            int   oi = si[threadIdx.x + s];
            if (ov > sv[threadIdx.x] ||
                (ov == sv[threadIdx.x] && oi < si[threadIdx.x])) {
                sv[threadIdx.x] = ov;
                si[threadIdx.x] = oi;
            }
        }
        __syncthreads();
    }

    const int sym = si[0];
    if (threadIdx.x == 0) symbols[(long)b * L + t] = (float)sym;

    const float* erow = embed + (long)sym * E;
    for (int e = threadIdx.x; e < E; e += 256)
        xout[(long)b * E + e] = erow[e];
}

extern "C" void kernel_launch(void* const* d_in, const int* in_sizes, int n_in,
                              void* d_out, int out_size, void* d_ws, size_t ws_size,
                              hipStream_t stream)
{
    (void)in_sizes; (void)n_in; (void)out_size; (void)ws_size;

    constexpr int B = 128, D_IN = 512, E = 512, H = 512, V = 32000, L = 16, NL = 2;

    const float* latent      = (const float*)d_in[0];
    const float* embed_table = (const float*)d_in[1];
    const float* proj1_W     = (const float*)d_in[2];
    const float* proj1_b     = (const float*)d_in[3];
    const float* proj2_W     = (const float*)d_in[4];
    const float* proj2_b     = (const float*)d_in[5];
    const float* rnn_Wih     = (const float*)d_in[6];
    const float* rnn_Whh     = (const float*)d_in[7];
    const float* rnn_bih     = (const float*)d_in[8];
    const float* rnn_bhh     = (const float*)d_in[9];
    const float* sos_embed   = (const float*)d_in[10];

    float* ws = (float*)d_ws;
    float* hA = ws;                                  // [NL][B][H] current hidden
    float* hB = hA + (size_t)NL * B * H;             // [NL][B][H] next hidden
    float* x  = hB + (size_t)NL * B * H;             // [B][E] current input

    float* out_sym = (float*)d_out;                  // [B][L]
    float* out_log = out_sym + (size_t)B * L;        // [B][L][V]

    const int blocksH = H / 64;                      // 8 blocks
    const int blocksV = V / 64;                      // 500 blocks

    // h0 = latent @ proj1_W^T + proj1_b  -> hA layer 0, then replicate to layer 1
    gemm_xwT_wmma<<<blocksH, 256, 0, stream>>>(
        latent, proj1_W, proj1_b, nullptr, nullptr, nullptr,
        hA, (long)H, H, D_IN, 0, /*tanh=*/0);
    hipMemcpyAsync(hA + (size_t)B * H, hA, sizeof(float) * (size_t)B * H,
                   hipMemcpyDeviceToDevice, stream);

    bcast_sos_kernel<<<(B * E + 255) / 256, 256, 0, stream>>>(sos_embed, x, B * E, E);

    for (int t = 0; t < L; ++t) {
        // Layer 0: hB0 = tanh(x @ Wih0^T + bih0 + hA0 @ Whh0^T + bhh0)
        gemm_xwT_wmma<<<blocksH, 256, 0, stream>>>(
            x, rnn_Wih, rnn_bih,
            hA, rnn_Whh, rnn_bhh,
            hB, (long)H, H, E, H, /*tanh=*/1);

        // Layer 1: hB1 = tanh(hB0 @ Wih1^T + bih1 + hA1 @ Whh1^T + bhh1)
        gemm_xwT_wmma<<<blocksH, 256, 0, stream>>>(
            hB, rnn_Wih + (size_t)H * E, rnn_bih + H,
            hA + (size_t)B * H, rnn_Whh + (size_t)H * H, rnn_bhh + H,
            hB + (size_t)B * H, (long)H, H, E, H, /*tanh=*/1);

        // logits[b][t][:] = hB1 @ proj2_W^T + proj2_b  (written straight into d_out)
        gemm_xwT_wmma<<<blocksV, 256, 0, stream>>>(
            hB + (size_t)B * H, proj2_W, proj2_b, nullptr, nullptr, nullptr,
            out_log + (size_t)t * V, (long)L * V, V, H, 0, /*tanh=*/0);

        // sym = argmax(logits); x = embed_table[sym]
        argmax_embed_kernel<<<B, 256, 0, stream>>>(
            out_log + (size_t)t * V, (long)L * V,
            out_sym, t, L, V, embed_table, E, x);

        // swap hidden-state double buffers
        float* tmp = hA; hA = hB; hB = tmp;
    }
}